// GNNRecommender_47760036331720
// MI455X (gfx1250) — compile-verified
//
#include <hip/hip_runtime.h>
#include <hip/hip_bf16.h>

typedef float v2f __attribute__((ext_vector_type(2)));
typedef float v8f __attribute__((ext_vector_type(8)));

// ---------------------------------------------------------------------------
// Degree / normalization kernels
// ---------------------------------------------------------------------------
__global__ void k_deg_init(float* __restrict__ deg, int n) {
    int i = blockIdx.x * blockDim.x + threadIdx.x;
    if (i < n) deg[i] = 1.0f;   // self-loop contributes 1 to every node's degree
}

__global__ void k_deg_count(const int* __restrict__ dst, float* __restrict__ deg, int ne) {
    int e = blockIdx.x * blockDim.x + threadIdx.x;
    if (e < ne) unsafeAtomicAdd(&deg[dst[e]], 1.0f);
}

__global__ void k_dinv(float* __restrict__ deg, int n) {
    int i = blockIdx.x * blockDim.x + threadIdx.x;
    if (i < n) {
        float d = deg[i];            // d >= 1 always (self-loops)
        deg[i] = rsqrtf(d);          // in-place: buffer becomes dinv
    }
}

// ---------------------------------------------------------------------------
// Dense GEMM via V_WMMA_F32_16X16X4_F32.  H[nrows,NOUT] = X[nrows,K] @ W[K,NOUT]
// One wave owns a 16-row tile and sweeps all NOUT/16 column tiles, K-loop of 4.
// A fragments are cached in registers across column tiles; W staged in LDS.
// ---------------------------------------------------------------------------
template <int K, int NOUT>
__global__ void __launch_bounds__(128) k_gemm_wmma(const float* __restrict__ X,
                                                   const float* __restrict__ W,
                                                   float* __restrict__ H,
                                                   int nrows) {
    __shared__ float wsh[K * NOUT];

    const int tid = threadIdx.x;
    // cooperative LDS fill of W (float4 transfers)
    {
        constexpr int TOT4 = (K * NOUT) / 4;
        const float4* W4 = reinterpret_cast<const float4*>(W);
        float4* S4 = reinterpret_cast<float4*>(wsh);
        for (int i = tid; i < TOT4; i += 128) S4[i] = W4[i];
    }
    __syncthreads();

    const int wave = tid >> 5;
    const int lane = tid & 31;
    const int m0 = (blockIdx.x * 4 + wave) * 16;
    if (m0 >= nrows) return;           // wave-uniform: EXEC stays all-1s for WMMA

    const int lmod = lane & 15;        // col (B/C/D) or row (A) within tile
    const int hi2  = (lane >> 4) * 2;  // K sub-offset for upper half-wave
    const int rowoff = (lane >> 4) * 8;

    // Preload all A fragments for this 16-row strip: K/4 steps, 2 f32 each lane.
    // Layout (ISA 7.12.2, 32-bit A 16x4): lane m = lmod, VGPR v holds K = v + 2*hi.
    v2f afrag[K / 4];
    const float* xrow = X + (size_t)(m0 + lmod) * K;
#pragma unroll
    for (int kk = 0; kk < K / 4; ++kk) {
        const float* p = xrow + kk * 4 + hi2;
        v2f a; a.x = p[0]; a.y = p[1];
        afrag[kk] = a;
    }

#pragma unroll
    for (int nt = 0; nt < NOUT / 16; ++nt) {
        const int n0 = nt * 16;
        v8f acc = {};
#pragma unroll
        for (int kk = 0; kk < K / 4; ++kk) {
            const int kb = kk * 4 + hi2;
            v2f b;
            b.x = wsh[(kb + 0) * NOUT + n0 + lmod];
            b.y = wsh[(kb + 1) * NOUT + n0 + lmod];
            acc = __builtin_amdgcn_wmma_f32_16x16x4_f32(
                /*neg_a=*/false, afrag[kk], /*neg_b=*/false, b,
                /*c_mod=*/(short)0, acc, /*reuse_a=*/false, /*reuse_b=*/false);
        }
        // C/D layout: VGPR r -> row m0 + r + 8*(lane>=16), col n0 + lmod
        float* orow = H + (size_t)(m0 + rowoff) * NOUT + n0 + lmod;
#pragma unroll
        for (int r = 0; r < 8; ++r) orow[(size_t)r * NOUT] = acc[r];
    }
}

// ---------------------------------------------------------------------------
// Aggregation: OUT = selfloop(h)*dinv^2 + bias, then scatter-add over edges.
// ---------------------------------------------------------------------------
template <int F>
__global__ void k_agg_init(const float* __restrict__ H,
                           const float* __restrict__ dinv,
                           const float* __restrict__ bias,
                           float* __restrict__ OUT, int n) {
    long long t = (long long)blockIdx.x * blockDim.x + threadIdx.x;
    if (t >= (long long)n * F) return;
    int i = (int)(t / F);
    int f = (int)(t % F);
    float di = dinv[i];
    OUT[t] = H[t] * di * di + bias[f];
}

template <int F>
__global__ void k_scatter(const float* __restrict__ H,
                          const int* __restrict__ src,
                          const int* __restrict__ dst,
                          const float* __restrict__ dinv,
                          float* __restrict__ OUT, long long nwork) {
    long long t = (long long)blockIdx.x * blockDim.x + threadIdx.x;
    if (t >= nwork) return;
    constexpr int CH = F / 4;                 // float4 chunks per edge
    int e = (int)(t / CH);
    int c = (int)(t % CH) * 4;
    int s = src[e], d = dst[e];
    float w = dinv[s] * dinv[d];
    const float4 hv = *reinterpret_cast<const float4*>(H + (size_t)s * F + c);
    float* o = OUT + (size_t)d * F + c;
    unsafeAtomicAdd(o + 0, hv.x * w);
    unsafeAtomicAdd(o + 1, hv.y * w);
    unsafeAtomicAdd(o + 2, hv.z * w);
    unsafeAtomicAdd(o + 3, hv.w * w);
}

__global__ void k_relu(float* __restrict__ v, long long n) {
    long long t = (long long)blockIdx.x * blockDim.x + threadIdx.x;
    if (t < n) v[t] = fmaxf(v[t], 0.0f);
}

// ---------------------------------------------------------------------------
extern "C" void kernel_launch(void* const* d_in, const int* in_sizes, int n_in,
                              void* d_out, int out_size, void* d_ws, size_t ws_size,
                              hipStream_t stream) {
    constexpr int D_IN = 128, D_HID = 128, D_OUT = 64;

    const float* x  = (const float*)d_in[0];
    const float* W1 = (const float*)d_in[1];
    const float* b1 = (const float*)d_in[2];
    const float* W2 = (const float*)d_in[3];
    const float* b2 = (const float*)d_in[4];
    const int*   ei = (const int*)d_in[5];

    const int N = in_sizes[0] / D_IN;
    const int E = in_sizes[5] / 2;
    const int* src = ei;        // edge_index[0]
    const int* dst = ei + E;    // edge_index[1]

    // workspace carve-up (256B aligned)
    auto align = [](size_t v) { return (v + 255) & ~(size_t)255; };
    char* ws = (char*)d_ws;
    size_t off = 0;
    float* dinv   = (float*)(ws + off); off = align(off + (size_t)N * sizeof(float));
    float* h1     = (float*)(ws + off); off = align(off + (size_t)N * D_HID * sizeof(float));
    float* h1agg  = (float*)(ws + off); off = align(off + (size_t)N * D_HID * sizeof(float));
    float* h2     = (float*)(ws + off); off = align(off + (size_t)N * D_OUT * sizeof(float));
    (void)ws_size;

    float* out = (float*)d_out;
    auto nb = [](long long n, int t) { return (unsigned)((n + t - 1) / t); };

    // --- degrees & symmetric norm --------------------------------------
    k_deg_init <<<nb(N, 256), 256, 0, stream>>>(dinv, N);
    k_deg_count<<<nb(E, 256), 256, 0, stream>>>(dst, dinv, E);
    k_dinv     <<<nb(N, 256), 256, 0, stream>>>(dinv, N);

    const int mtiles = (N + 15) / 16;
    const unsigned gemm_blocks = (unsigned)((mtiles + 3) / 4);

    // --- layer 1: h1 = x @ W1 ; agg ; +b1 ; relu -----------------------
    k_gemm_wmma<D_IN, D_HID><<<gemm_blocks, 128, 0, stream>>>(x, W1, h1, N);
    k_agg_init<D_HID><<<nb((long long)N * D_HID, 256), 256, 0, stream>>>(h1, dinv, b1, h1agg, N);
    k_scatter<D_HID><<<nb((long long)E * (D_HID / 4), 256), 256, 0, stream>>>(
        h1, src, dst, dinv, h1agg, (long long)E * (D_HID / 4));
    k_relu<<<nb((long long)N * D_HID, 256), 256, 0, stream>>>(h1agg, (long long)N * D_HID);

    // --- layer 2: h2 = relu @ W2 ; agg ; +b2 --> d_out -----------------
    k_gemm_wmma<D_HID, D_OUT><<<gemm_blocks, 128, 0, stream>>>(h1agg, W2, h2, N);
    k_agg_init<D_OUT><<<nb((long long)N * D_OUT, 256), 256, 0, stream>>>(h2, dinv, b2, out, N);
    k_scatter<D_OUT><<<nb((long long)E * (D_OUT / 4), 256), 256, 0, stream>>>(
        h2, src, dst, dinv, out, (long long)E * (D_OUT / 4));
}